// TwoStageAttentionLayer_27693949125223
// MI455X (gfx1250) — compile-verified
//
#include <hip/hip_runtime.h>
#include <math.h>

// ---------------------------------------------------------------------------
// TwoStageAttentionLayer forward for MI455X (gfx1250).
// B=2 N=32 D=8 L=16 C=128 H=8 FACTOR=10 D_FF=512 K=3 DM2=2048
// All GEMMs run through a fp32 WMMA (v_wmma_f32_16x16x4_f32) tiled kernel:
// one wave computes a 32x32 output tile (4 accumulators, shared A/B fragments).
// ---------------------------------------------------------------------------

#define Bv   2
#define Nv   32
#define Dv   8
#define Lv   16
#define Cv   128
#define Hv   8
#define FACv 10
#define DFF  512
#define Kv   3
#define DM2v 2048
#define T1   (Bv*Nv*Dv*Lv)      // 8192 tokens of C
#define EPSv 1e-5f

typedef __attribute__((ext_vector_type(2))) float v2f;
typedef __attribute__((ext_vector_type(8))) float v8f;

__device__ __forceinline__ float gelu_f(float x) {
    return 0.5f * x * (1.0f + erff(x * 0.70710678118654752440f));
}

// ---------------------------------------------------------------------------
// WMMA fp32 GEMM: out[M,N] = act(A[M,K] @ W[K,N] + bias[N])
// One wave -> 32x32 output tile: 2 A-fragments x 2 B-fragments -> 4 WMMAs/k-step.
// Requires: M%32==0, N%32==0, K%4==0. act: 0=none, 1=exact gelu.
// ---------------------------------------------------------------------------
__global__ __launch_bounds__(256) void gemm_wmma_kernel(
    const float* __restrict__ A, const float* __restrict__ W,
    const float* __restrict__ bias, float* __restrict__ out,
    int M, int N, int K, int act, int totalTiles)
{
    int wave = blockIdx.x * (blockDim.x >> 5) + (threadIdx.x >> 5);
    if (wave >= totalTiles) return;              // uniform per wave: EXEC stays all-ones
    int nT = N >> 5;
    int tm = wave / nT, tn = wave % nT;
    int m0 = tm << 5, n0 = tn << 5;

    int lane = threadIdx.x & 31;
    int r16  = lane & 15;
    int kh   = lane >> 4;                        // 0 or 1 (half-wave K offset)

    const float* Arow0 = A + (size_t)(m0 + r16) * K;        // rows m0..m0+15
    const float* Arow1 = Arow0 + (size_t)16 * K;            // rows m0+16..m0+31

    v8f c00 = {};   // (m0..m0+15)  x (n0..n0+15)
    v8f c01 = {};   // (m0..m0+15)  x (n0+16..n0+31)
    v8f c10 = {};   // (m0+16..31)  x (n0..n0+15)
    v8f c11 = {};   // (m0+16..31)  x (n0+16..n0+31)

    for (int k = 0; k < K; k += 4) {
        int kk = k + (kh << 1);                  // K index = 2*kh + vgpr
        v2f a0; a0.x = Arow0[kk]; a0.y = Arow0[kk + 1];
        v2f a1; a1.x = Arow1[kk]; a1.y = Arow1[kk + 1];
        const float* W0 = W + (size_t)kk * N + n0 + r16;
        const float* W1 = W0 + N;
        v2f b0; b0.x = W0[0];  b0.y = W1[0];
        v2f b1; b1.x = W0[16]; b1.y = W1[16];
        c00 = __builtin_amdgcn_wmma_f32_16x16x4_f32(false, a0, false, b0,
                                                    (short)0, c00, false, false);
        c01 = __builtin_amdgcn_wmma_f32_16x16x4_f32(false, a0, false, b1,
                                                    (short)0, c01, false, false);
        c10 = __builtin_amdgcn_wmma_f32_16x16x4_f32(false, a1, false, b0,
                                                    (short)0, c10, false, false);
        c11 = __builtin_amdgcn_wmma_f32_16x16x4_f32(false, a1, false, b1,
                                                    (short)0, c11, false, false);
    }

    int row0 = m0 + (kh << 3);                   // C/D: vgpr r -> M = r + 8*kh
    int col  = n0 + r16;
    float bb0 = bias[col];
    float bb1 = bias[col + 16];
#pragma unroll
    for (int r = 0; r < 8; ++r) {
        float v00 = c00[r] + bb0;
        float v01 = c01[r] + bb1;
        float v10 = c10[r] + bb0;
        float v11 = c11[r] + bb1;
        if (act) {
            v00 = gelu_f(v00); v01 = gelu_f(v01);
            v10 = gelu_f(v10); v11 = gelu_f(v11);
        }
        out[(size_t)(row0 + r) * N + col]           = v00;
        out[(size_t)(row0 + r) * N + col + 16]      = v01;
        out[(size_t)(row0 + 16 + r) * N + col]      = v10;
        out[(size_t)(row0 + 16 + r) * N + col + 16] = v11;
    }
}

// ---------------------------------------------------------------------------
// Small-sequence attention: one wave (32 lanes) per (problem, head, q-row).
// Q rows: [(qPeriod? p%qPeriod : p)*Sq + sq], K/V rows: [p*Sk + sk], dm = H*dk.
// ---------------------------------------------------------------------------
__global__ __launch_bounds__(32) void attn_kernel(
    const float* __restrict__ Q, const float* __restrict__ Kt,
    const float* __restrict__ Vt, float* __restrict__ O,
    int P, int Sq, int Sk, int Hh, int dk, int qPeriod, float scale)
{
    __shared__ float sc[32];
    int bid = blockIdx.x;
    int sq = bid % Sq; bid /= Sq;
    int h  = bid % Hh; bid /= Hh;
    int p  = bid;
    int dm = Hh * dk;
    int qp = (qPeriod > 0) ? (p % qPeriod) : p;

    const float* q = Q + ((size_t)qp * Sq + sq) * dm + h * dk;
    int lane = threadIdx.x;

    float s = -3.0e38f;
    if (lane < Sk) {
        const float* kr = Kt + ((size_t)p * Sk + lane) * dm + h * dk;
        float acc = 0.f;
        for (int i = 0; i < dk; ++i) acc += q[i] * kr[i];
        s = acc * scale;
    }
    float m = s;
    for (int off = 16; off; off >>= 1) m = fmaxf(m, __shfl_xor(m, off, 32));
    float e = (lane < Sk) ? __expf(s - m) : 0.f;
    float sum = e;
    for (int off = 16; off; off >>= 1) sum += __shfl_xor(sum, off, 32);
    sc[lane] = e / sum;
    __syncthreads();

    float* o = O + ((size_t)p * Sq + sq) * dm + h * dk;
    const float* vb = Vt + (size_t)p * Sk * dm + h * dk;
    for (int d = lane; d < dk; d += 32) {
        float acc = 0.f;
        for (int sk = 0; sk < Sk; ++sk) acc += sc[sk] * vb[(size_t)sk * dm + d];
        o[d] = acc;
    }
}

// ---------------------------------------------------------------------------
// out[t,:] = LN(a[t,:] + b[t,:]) * g + beta   (C = blockDim.x = 128)
// ---------------------------------------------------------------------------
__global__ __launch_bounds__(Cv) void ln_res_kernel(
    const float* __restrict__ a, const float* __restrict__ b,
    const float* __restrict__ g, const float* __restrict__ be,
    float* __restrict__ out)
{
    __shared__ float s1[Cv], s2[Cv];
    int t = blockIdx.x, i = threadIdx.x;
    float v = a[(size_t)t * Cv + i] + b[(size_t)t * Cv + i];
    s1[i] = v; s2[i] = v * v;
    __syncthreads();
    for (int st = Cv >> 1; st > 0; st >>= 1) {
        if (i < st) { s1[i] += s1[i + st]; s2[i] += s2[i + st]; }
        __syncthreads();
    }
    float mean = s1[0] * (1.0f / Cv);
    float var  = s2[0] * (1.0f / Cv) - mean * mean;
    float ir = rsqrtf(var + EPSv);
    out[(size_t)t * Cv + i] = (v - mean) * ir * g[i] + be[i];
}

// in: [outer, A, Bd, C] -> out: [outer, Bd, A, C]
__global__ void permute_kernel(const float* __restrict__ in, float* __restrict__ out,
                               int outer, int A, int Bd, int C, int total)
{
    int i = blockIdx.x * blockDim.x + threadIdx.x;
    if (i >= total) return;
    int c = i % C; int r = i / C;
    int b = r % Bd; r /= Bd;
    int a = r % A;  int o = r / A;
    out[(((size_t)o * Bd + b) * A + a) * C + c] = in[i];
}

// Kv[((b*N+n)*K+k)*D+d, :] = Dq[(b*N + topk[b,n,k])*D+d, :]   (rows of DM2)
__global__ void gather_kernel(const float* __restrict__ Dq, const int* __restrict__ topk,
                              float* __restrict__ Kvb, int total)
{
    int i = blockIdx.x * blockDim.x + threadIdx.x;
    if (i >= total) return;
    int c = i % DM2v; int r = i / DM2v;
    int d = r % Dv; r /= Dv;
    int k = r % Kv; r /= Kv;
    int n = r % Nv; int b = r / Nv;
    int idx = topk[(b * Nv + n) * Kv + k];
    Kvb[i] = Dq[(((size_t)(b * Nv + idx) * Dv + d) * DM2v) + c];
}

__global__ void add_kernel(const float* __restrict__ a, const float* __restrict__ b,
                           float* __restrict__ out, int total)
{
    int i = blockIdx.x * blockDim.x + threadIdx.x;
    if (i < total) out[i] = a[i] + b[i];
}

// ---------------------------------------------------------------------------
static inline void gemm(const float* A, const float* W, const float* bias, float* out,
                        int M, int N, int K, int act, hipStream_t s)
{
    int tiles = (M / 32) * (N / 32);
    int blocks = (tiles + 7) / 8;
    gemm_wmma_kernel<<<blocks, 256, 0, s>>>(A, W, bias, out, M, N, K, act, tiles);
}

extern "C" void kernel_launch(void* const* d_in, const int* in_sizes, int n_in,
                              void* d_out, int out_size, void* d_ws, size_t ws_size,
                              hipStream_t stream)
{
    (void)in_sizes; (void)n_in; (void)out_size; (void)ws_size;
    const float* x    = (const float*)d_in[0];
    const int*   topk = (const int*)d_in[1];
#define PR(i) ((const float*)d_in[i])
    // param indices: 2..9 ta_{wq,bq,wk,bk,wv,bv,wo,bo}; 10..17 ds; 18..25 dr; 26..33 ca;
    // 34 router; 35..42 n1..n4 g/b; 43..46 m1; 47..50 m2; 51,52 cn; 53..56 cm.

    float* ws = (float*)d_ws;
    // Arena (floats). Peak ~16.6M floats (~63 MB).
    float* bufA = ws;                        // 4,194,304  (ff 8192x512 / Vc / ff2)
    float* b1   = bufA + 4194304;            // 3,145,728  (K1/K2/K3/Kv/...)
    float* b2   = b1 + 3145728;              // 3,145,728  (V1/V2/V3/Kc)
    float* b3   = b2 + 3145728;              // 1,310,720  (Q1/Q3/Qc)
    float* b4   = b3 + 1310720;              // 1,310,720  (O1/O2/O3/Oc)
    float* b5   = b4 + 1310720;              // 1,310,720  (proj/buf/rec/ctx)
    float* b6   = b5 + 1310720;              // 1,048,576  (t / d_in / d2 / fused)
    float* b7   = b6 + 1048576;              // 1,048,576  (t2 / d / Dq)
    float* rQ   = b7 + 1048576;              // 20,480     (routerQ)

    const int TC  = T1 * Cv;                 // 1,048,576 elements
    const int EB  = 256;

    // ---- Stage 1: time attention over L (per (b,n,d)) ----
    gemm(x, PR(2), PR(3), b3, T1, Cv, Cv, 0, stream);      // Q1
    gemm(x, PR(4), PR(5), b1, T1, Cv, Cv, 0, stream);      // K1
    gemm(x, PR(6), PR(7), b2, T1, Cv, Cv, 0, stream);      // V1
    attn_kernel<<<512 * Hv * Lv, 32, 0, stream>>>(b3, b1, b2, b4,
        512, Lv, Lv, Hv, 16, 0, 0.25f);
    gemm(b4, PR(8), PR(9), b5, T1, Cv, Cv, 0, stream);     // proj
    ln_res_kernel<<<T1, Cv, 0, stream>>>(x, b5, PR(35), PR(36), b6);   // t (n1)

    // ---- MLP1 + n2 ----
    gemm(b6, PR(43), PR(44), bufA, T1, DFF, Cv, 1, stream);            // gelu(t@w1+b1)
    gemm(bufA, PR(45), PR(46), b5, T1, Cv, DFF, 0, stream);
    ln_res_kernel<<<T1, Cv, 0, stream>>>(b6, b5, PR(37), PR(38), b7);  // t2

    // ---- d_in = transpose (D,L) -> (L,D) ----
    permute_kernel<<<(TC + EB - 1) / EB, EB, 0, stream>>>(b7, b6, Bv * Nv, Dv, Lv, Cv, TC);

    // ---- ds: router attends to d_in (per (b,n,l): Sq=10, Sk=8) ----
    gemm(PR(34), PR(10), PR(11), rQ, Lv * FACv, Cv, Cv, 0, stream);    // routerQ [160,128]
    gemm(b6, PR(12), PR(13), b1, T1, Cv, Cv, 0, stream);               // K2
    gemm(b6, PR(14), PR(15), b2, T1, Cv, Cv, 0, stream);               // V2
    attn_kernel<<<1024 * Hv * FACv, 32, 0, stream>>>(rQ, b1, b2, b4,
        1024, FACv, Dv, Hv, 16, Lv, 0.25f);
    gemm(b4, PR(16), PR(17), b5, 1024 * FACv, Cv, Cv, 0, stream);      // buf [10240,128]

    // ---- dr: d_in attends to buf (per (b,n,l): Sq=8, Sk=10) ----
    gemm(b6, PR(18), PR(19), b3, T1, Cv, Cv, 0, stream);               // Q3
    gemm(b5, PR(20), PR(21), b1, 1024 * FACv, Cv, Cv, 0, stream);      // K3
    gemm(b5, PR(22), PR(23), b2, 1024 * FACv, Cv, Cv, 0, stream);      // V3
    attn_kernel<<<1024 * Hv * Dv, 32, 0, stream>>>(b3, b1, b2, b4,
        1024, Dv, FACv, Hv, 16, 0, 0.25f);
    gemm(b4, PR(24), PR(25), b5, T1, Cv, Cv, 0, stream);               // rec
    ln_res_kernel<<<T1, Cv, 0, stream>>>(b6, b5, PR(39), PR(40), b7);  // d (n3)

    // ---- MLP2 + n4 ----
    gemm(b7, PR(47), PR(48), bufA, T1, DFF, Cv, 1, stream);
    gemm(bufA, PR(49), PR(50), b5, T1, Cv, DFF, 0, stream);
    ln_res_kernel<<<T1, Cv, 0, stream>>>(b7, b5, PR(41), PR(42), b6);  // d2

    // ---- back-transpose (L,D) -> (D,L); gather top-k neighbors ----
    permute_kernel<<<(TC + EB - 1) / EB, EB, 0, stream>>>(b6, b7, Bv * Nv, Lv, Dv, Cv, TC);
    {
        int tot = Bv * Nv * Kv * Dv * DM2v;                            // 3,145,728
        gather_kernel<<<(tot + EB - 1) / EB, EB, 0, stream>>>(b7, topk, b1, tot);
    }

    // ---- cross attention over DM2=2048 (per (b,n): Sq=8, Sk=24, dk=256) ----
    gemm(b7, PR(26), PR(27), b3, 512, DM2v, DM2v, 0, stream);          // Qc
    gemm(b1, PR(28), PR(29), b2, 1536, DM2v, DM2v, 0, stream);         // Kc
    gemm(b1, PR(30), PR(31), bufA, 1536, DM2v, DM2v, 0, stream);       // Vc
    attn_kernel<<<64 * Hv * Dv, 32, 0, stream>>>(b3, b2, bufA, b4,
        64, Dv, Kv * Dv, Hv, 256, 0, 0.0625f);
    gemm(b4, PR(32), PR(33), b5, 512, DM2v, DM2v, 0, stream);          // ctx
    ln_res_kernel<<<T1, Cv, 0, stream>>>(b7, b5, PR(51), PR(52), b6);  // fused (cn)

    // ---- cm MLP + final residual ----
    gemm(b6, PR(53), PR(54), bufA, T1, 2 * Cv, Cv, 1, stream);
    gemm(bufA, PR(55), PR(56), b5, T1, Cv, 2 * Cv, 0, stream);
    add_kernel<<<(TC + EB - 1) / EB, EB, 0, stream>>>(b6, b5, (float*)d_out, TC);
#undef PR
}